// Model_19456201851455
// MI455X (gfx1250) — compile-verified
//
#include <hip/hip_runtime.h>
#include <math.h>

typedef __attribute__((ext_vector_type(2))) float v2f;
typedef __attribute__((ext_vector_type(8))) float v8f;

#define BATCH  8
#define SEQ    4096
#define IND    32
#define HID    256
#define OUTD   32
#define CH_T   64
#define NCHUNK (SEQ / CH_T)   // 64

// ---- fp32 WMMA wrapper: D = A(16x4) * B(4x16) + C --------------------------
__device__ __forceinline__ v8f wmma_f32(v2f a, v2f b, v8f c) {
    return __builtin_amdgcn_wmma_f32_16x16x4_f32(false, a, false, b,
                                                 (short)0, c, false, false);
}

// lambda = exp(-e^nu) * exp(i*e^theta), via HW trans ops
__device__ __forceinline__ void lam_params(const float* __restrict__ nu_log,
                                           const float* __restrict__ theta_log,
                                           int h, float& lre, float& lim) {
    float mag = __expf(-__expf(nu_log[h]));
    float ph  = __expf(theta_log[h]);
    float s, c;
    __sincosf(ph, &s, &c);
    lre = mag * c;
    lim = mag * s;
}

// Compute F = (U_chunk @ B^T) * gamma into Flds via WMMA.
// Ulds: [CH_T][IND], Blds: [HID][IND], Flds: [CH_T][HID], Glds: gamma[HID]
__device__ __forceinline__ void compute_F_wmma(const float* __restrict__ Ulds,
                                               const float* __restrict__ Blds,
                                               float* __restrict__ Flds,
                                               const float* __restrict__ Glds,
                                               int lane, int wave) {
    const int m_l   = lane & 15;
    const int khalf = (lane >> 4) << 1;            // 0 or 2
    const int MT = CH_T / 16;                      // 4
    const int NT = HID / 16;                       // 16
    for (int tile = wave; tile < MT * NT; tile += 8) {
        int mt = tile % MT, nt = tile / MT;
        int m = mt * 16 + m_l;
        int n = nt * 16 + m_l;
        v8f acc = {};
        #pragma unroll
        for (int kk = 0; kk < IND / 4; ++kk) {
            int k0 = kk * 4 + khalf;
            v2f a, b;
            a.x = Ulds[m * IND + k0];
            a.y = Ulds[m * IND + k0 + 1];
            b.x = Blds[n * IND + k0];
            b.y = Blds[n * IND + k0 + 1];
            acc = wmma_f32(a, b, acc);
        }
        float gam = Glds[n];                       // precomputed, no trans ops here
        int rbase = mt * 16 + ((lane >> 4) << 3);
        #pragma unroll
        for (int v = 0; v < 8; ++v)
            Flds[(rbase + v) * HID + n] = acc[v] * gam;
    }
}

// ---- Phase 1: per-chunk local scan from zero -> carry-out ------------------
__global__ __launch_bounds__(256)
void lru_chunk_carry(const float* __restrict__ u,
                     const float* __restrict__ nu_log,
                     const float* __restrict__ theta_log,
                     const float* __restrict__ Bm,
                     float* __restrict__ s_out) {
    extern __shared__ float smem[];
    float* Ulds = smem;                       // 2048
    float* Blds = Ulds + CH_T * IND;          // 8192
    float* Flds = Blds + HID * IND;           // 16384
    float* Glds = Flds + CH_T * HID;          // 256

    const int b    = blockIdx.x / NCHUNK;
    const int c    = blockIdx.x % NCHUNK;
    const int tid  = threadIdx.x;
    const int lane = tid & 31;
    const int wave = tid >> 5;

    // per-thread recurrence params (one h per thread), and gamma table
    const int h = tid;
    float lre, lim;
    lam_params(nu_log, theta_log, h, lre, lim);
    {
        float mag = __expf(-__expf(nu_log[h]));
        Glds[h] = sqrtf(1.0f - mag * mag + 1e-8f);
    }

    const float* ug = u + ((size_t)b * SEQ + (size_t)c * CH_T) * IND;
    for (int i = tid; i < CH_T * IND; i += 256) Ulds[i] = ug[i];
    for (int i = tid; i < HID * IND;  i += 256) Blds[i] = Bm[i];
    __syncthreads();

    compute_F_wmma(Ulds, Blds, Flds, Glds, lane, wave);
    __syncthreads();

    // local scan from zero state
    float xr = 0.f, xi = 0.f;
    for (int t = 0; t < CH_T; ++t) {
        float f  = Flds[t * HID + h];
        float nr = lre * xr - lim * xi + f;
        float ni = lre * xi + lim * xr;
        xr = nr; xi = ni;
    }
    size_t idx = ((size_t)(b * NCHUNK + c) * HID + h) * 2;
    s_out[idx]     = xr;
    s_out[idx + 1] = xi;
}

// ---- Phase 2: inter-chunk scan (64 serial steps per batch) -----------------
__global__ __launch_bounds__(256)
void lru_carry_scan(const float* __restrict__ nu_log,
                    const float* __restrict__ theta_log,
                    const float* __restrict__ s_in,
                    float* __restrict__ carry_out) {
    const int b = blockIdx.x;
    const int h = threadIdx.x;
    // lambda^CH_T = exp(-CH_T*e^nu) * exp(i*CH_T*e^theta)
    float magT = __expf(-(float)CH_T * __expf(nu_log[h]));
    float angT = (float)CH_T * __expf(theta_log[h]);
    float s, c2;
    __sincosf(angT, &s, &c2);
    float lre = magT * c2;
    float lim = magT * s;
    float xr = 0.f, xi = 0.f;
    for (int c = 0; c < NCHUNK; ++c) {
        size_t idx = ((size_t)(b * NCHUNK + c) * HID + h) * 2;
        carry_out[idx]     = xr;     // state entering chunk c
        carry_out[idx + 1] = xi;
        float sr = s_in[idx], si = s_in[idx + 1];
        float nr = lre * xr - lim * xi + sr;
        float ni = lre * xi + lim * xr + si;
        xr = nr; xi = ni;
    }
}

// ---- Phase 3: rescan from carry-in, then Y = Xr@C^T + U@D^T ----------------
__global__ __launch_bounds__(256)
void lru_final(const float* __restrict__ u,
               const float* __restrict__ nu_log,
               const float* __restrict__ theta_log,
               const float* __restrict__ Bm,
               const float* __restrict__ Cm,
               const float* __restrict__ Dm,
               const float* __restrict__ carry_in,
               float* __restrict__ out) {
    extern __shared__ float smem[];
    float* Ulds = smem;                       // 2048
    float* Blds = Ulds + CH_T * IND;          // 8192
    float* Flds = Blds + HID * IND;           // 16384  (F, then Re(x))
    float* Glds = Flds + CH_T * HID;          // 256
    float* Clds = Glds + HID;                 // 8192   C: [OUTD][HID]

    const int b    = blockIdx.x / NCHUNK;
    const int c    = blockIdx.x % NCHUNK;
    const int tid  = threadIdx.x;
    const int lane = tid & 31;
    const int wave = tid >> 5;

    const int h = tid;
    float lre, lim;
    lam_params(nu_log, theta_log, h, lre, lim);
    {
        float mag = __expf(-__expf(nu_log[h]));
        Glds[h] = sqrtf(1.0f - mag * mag + 1e-8f);
    }

    const float* ug = u + ((size_t)b * SEQ + (size_t)c * CH_T) * IND;
    for (int i = tid; i < CH_T * IND;  i += 256) Ulds[i] = ug[i];
    for (int i = tid; i < HID * IND;   i += 256) Blds[i] = Bm[i];
    for (int i = tid; i < OUTD * HID;  i += 256) Clds[i] = Cm[i];
    __syncthreads();

    compute_F_wmma(Ulds, Blds, Flds, Glds, lane, wave);
    __syncthreads();

    // scan from the chunk's incoming global state; overwrite Flds with Re(x)
    size_t cidx = ((size_t)(b * NCHUNK + c) * HID + h) * 2;
    float xr = carry_in[cidx], xi = carry_in[cidx + 1];
    for (int t = 0; t < CH_T; ++t) {
        float f  = Flds[t * HID + h];
        float nr = lre * xr - lim * xi + f;
        float ni = lre * xi + lim * xr;
        xr = nr; xi = ni;
        Flds[t * HID + h] = xr;               // in-place: read slot then write
    }
    __syncthreads();

    // Y tile per wave: MT=4 time-tiles x NT=2 out-tiles = 8 tiles = 8 waves
    const int m_l   = lane & 15;
    const int khalf = (lane >> 4) << 1;
    const int mt = wave & 3;
    const int nt = wave >> 2;
    const int m  = mt * 16 + m_l;             // local time row
    const int n  = nt * 16 + m_l;             // output channel
    v8f acc = {};
    #pragma unroll 8
    for (int kk = 0; kk < HID / 4; ++kk) {    // Xr @ C^T   (K = 256)
        int k0 = kk * 4 + khalf;
        v2f a, bb;
        a.x  = Flds[m * HID + k0];
        a.y  = Flds[m * HID + k0 + 1];
        bb.x = Clds[n * HID + k0];
        bb.y = Clds[n * HID + k0 + 1];
        acc  = wmma_f32(a, bb, acc);
    }
    #pragma unroll
    for (int kk = 0; kk < IND / 4; ++kk) {    // + U @ D^T  (K = 32)
        int k0 = kk * 4 + khalf;
        v2f a, bb;
        a.x  = Ulds[m * IND + k0];
        a.y  = Ulds[m * IND + k0 + 1];
        bb.x = Dm[n * IND + k0];
        bb.y = Dm[n * IND + k0 + 1];
        acc  = wmma_f32(a, bb, acc);
    }
    float* og = out + ((size_t)b * SEQ + (size_t)c * CH_T) * OUTD;
    int rbase = mt * 16 + ((lane >> 4) << 3);
    #pragma unroll
    for (int v = 0; v < 8; ++v)
        og[(rbase + v) * OUTD + n] = acc[v];
}

extern "C" void kernel_launch(void* const* d_in, const int* in_sizes, int n_in,
                              void* d_out, int out_size, void* d_ws, size_t ws_size,
                              hipStream_t stream) {
    const float* u         = (const float*)d_in[0];
    const float* nu_log    = (const float*)d_in[1];
    const float* theta_log = (const float*)d_in[2];
    const float* Bm        = (const float*)d_in[3];
    const float* Cm        = (const float*)d_in[4];
    const float* Dm        = (const float*)d_in[5];
    float* out = (float*)d_out;

    float* s_buf = (float*)d_ws;                              // [B][NC][H][2]
    float* carry = s_buf + (size_t)BATCH * NCHUNK * HID * 2;  // [B][NC][H][2]

    const size_t sh1 = (size_t)(CH_T * IND + HID * IND + CH_T * HID + HID)
                       * sizeof(float);                       // ~105 KB
    const size_t sh3 = sh1 + (size_t)(OUTD * HID) * sizeof(float); // ~137 KB

    lru_chunk_carry<<<BATCH * NCHUNK, 256, sh1, stream>>>(
        u, nu_log, theta_log, Bm, s_buf);
    lru_carry_scan<<<BATCH, HID, 0, stream>>>(
        nu_log, theta_log, s_buf, carry);
    lru_final<<<BATCH * NCHUNK, 256, sh3, stream>>>(
        u, nu_log, theta_log, Bm, Cm, Dm, carry, out);
}